// ContrastiveLearningLoss_68728066671109
// MI455X (gfx1250) — compile-verified
//
#include <hip/hip_runtime.h>
#include <hip/hip_bf16.h>
#include <math.h>

// CDNA5 / gfx1250, wave32.
// Fused contrastive (supervised InfoNCE-style) loss:
//   en = l2norm(e) * (1/sqrt(T))      -> dot(en_i, en_j) == sim_ij / T
//   per row i: logZ_i via sum of exp(shifted logits), positives via label match
//   loss = mean_i [ cnt_i>0 ? (cnt_i*log(sumexp_i) - possum_i)/cnt_i : 0 ]
// (shift by 1/T cancels exactly; see derivation in analysis)

typedef __attribute__((ext_vector_type(2))) float v2f;
typedef __attribute__((ext_vector_type(8))) float v8f;

#define DIMK   256          // embedding dim
#define KV2    128          // DIMK/2 float2 chunks per row
#define KSTEPS 64           // DIMK/4 wmma K-steps
#define TEMP   0.07f
#define WAVES  8            // waves per block in main kernel

// ---------------------------------------------------------------------------
// Kernel 1: row L2-normalize (+ 1/sqrt(T) scale) into workspace, zero d_out.
// One wave per row: 32 lanes x 8 floats (two float4 loads) = 256 floats.
// ---------------------------------------------------------------------------
__global__ void cl_normalize(const float* __restrict__ e,
                             float* __restrict__ en,
                             float* __restrict__ out,
                             int B)
{
    if (blockIdx.x == 0 && threadIdx.x == 0) out[0] = 0.0f;

    const int lane = threadIdx.x & 31;
    const int wave = threadIdx.x >> 5;
    const int row  = blockIdx.x * WAVES + wave;
    if (row >= B) return;

    const float4* src = (const float4*)(e + (size_t)row * DIMK);
    float4 a = src[lane];        // bytes [lane*16)        -> global_load_b128
    float4 b = src[lane + 32];   // second half of the row

    float ss = a.x * a.x + a.y * a.y + a.z * a.z + a.w * a.w
             + b.x * b.x + b.y * b.y + b.z * b.z + b.w * b.w;
    #pragma unroll
    for (int m = 1; m < 32; m <<= 1)
        ss += __shfl_xor(ss, m, 32);

    const float invSqrtT = 3.7796447300922722f;      // 1/sqrt(0.07)
    const float nrm = sqrtf(ss);
    const float s   = invSqrtT / fmaxf(nrm, 1e-12f);

    float4* dst = (float4*)(en + (size_t)row * DIMK);
    a.x *= s; a.y *= s; a.z *= s; a.w *= s;
    b.x *= s; b.y *= s; b.z *= s; b.w *= s;
    dst[lane]      = a;
    dst[lane + 32] = b;
}

// ---------------------------------------------------------------------------
// Kernel 2: fused sim-GEMM (v_wmma_f32_16x16x4_f32) + log-softmax stats +
// positive-mask accumulation. One wave owns a 16-row tile; the 16x256 A tile
// lives entirely in VGPRs (v2f[64]) and is reused across all column tiles.
// ---------------------------------------------------------------------------
__global__ void __launch_bounds__(32 * WAVES)
cl_loss(const float* __restrict__ en,
        const int*   __restrict__ labels,
        float*       __restrict__ out,
        int B)
{
    const int lane  = threadIdx.x & 31;
    const int wave  = threadIdx.x >> 5;
    const int tileA = blockIdx.x * WAVES + wave;     // 16-row tile index
    const int rowbase = tileA * 16;
    const int half  = lane >> 4;                     // 0: rows m=0..7, 1: m=8..15
    const int n16   = lane & 15;                     // column-within-tile / row-of-A

    const v2f* en2 = (const v2f*)en;

    // --- A tile resident in registers: lane covers row (rowbase+n16),
    //     K-pair (4k + 2*half .. +1)  [matches 16x4 f32 A operand layout] ---
    v2f areg[KSTEPS];
    {
        const int arow = rowbase + n16;
        const v2f* ap = en2 + (size_t)arow * KV2 + half;
        #pragma unroll
        for (int k = 0; k < KSTEPS; ++k)
            areg[k] = ap[2 * k];
    }

    // labels of my 8 rows (m = half*8 + r)
    int labRow[8];
    #pragma unroll
    for (int r = 0; r < 8; ++r)
        labRow[r] = labels[rowbase + half * 8 + r];

    float sumexp[8], possum[8], cnt[8];
    #pragma unroll
    for (int r = 0; r < 8; ++r) { sumexp[r] = 0.0f; possum[r] = 0.0f; cnt[r] = 0.0f; }

    const float SHIFT = 1.0f / TEMP;                 // max possible logit
    const int NT = B >> 4;

    for (int j = 0; j < NT; ++j) {
        const int col    = j * 16 + n16;
        const int labCol = labels[col];

        // two accumulator chains to break the WMMA RAW dependency
        v8f c0 = {}; v8f c1 = {};
        const v2f* bp = en2 + (size_t)col * KV2 + half;
        #pragma unroll
        for (int k = 0; k < KSTEPS; k += 2) {
            v2f b0 = bp[2 * k];
            v2f b1 = bp[2 * k + 2];
            c0 = __builtin_amdgcn_wmma_f32_16x16x4_f32(
                     false, areg[k],     false, b0, (short)0, c0, false, false);
            c1 = __builtin_amdgcn_wmma_f32_16x16x4_f32(
                     false, areg[k + 1], false, b1, (short)0, c1, false, false);
        }

        // epilogue: lane holds 8 logits, rows m=half*8+r, column n16 (branchless)
        #pragma unroll
        for (int r = 0; r < 8; ++r) {
            const float s = (c0[r] + c1[r]) - SHIFT;       // <= 0 always
            sumexp[r] += __expf(s);
            const int grow = rowbase + half * 8 + r;
            const float msk = (labCol == labRow[r] && col != grow) ? 1.0f : 0.0f;
            possum[r] = fmaf(msk, s, possum[r]);
            cnt[r]   += msk;
        }
    }

    // reduce the 16 column-lanes of each half (rows are disjoint across halves)
    #pragma unroll
    for (int m = 1; m < 16; m <<= 1) {
        #pragma unroll
        for (int r = 0; r < 8; ++r) {
            sumexp[r] += __shfl_xor(sumexp[r], m, 32);
            possum[r] += __shfl_xor(possum[r], m, 32);
            cnt[r]    += __shfl_xor(cnt[r],    m, 32);
        }
    }

    if (n16 == 0) {            // lane 0 -> rows 0..7, lane 16 -> rows 8..15
        float acc = 0.0f;
        #pragma unroll
        for (int r = 0; r < 8; ++r) {
            // shift cancels: loss_i = (cnt*log(sumexp) - possum)/cnt
            const float li = (cnt[r] * __logf(sumexp[r]) - possum[r]) / cnt[r];
            acc += (cnt[r] > 0.0f) ? li : 0.0f;
        }
        atomicAdd(out, acc / (float)B);
    }
}

// ---------------------------------------------------------------------------
extern "C" void kernel_launch(void* const* d_in, const int* in_sizes, int n_in,
                              void* d_out, int out_size, void* d_ws, size_t ws_size,
                              hipStream_t stream)
{
    const float* emb    = (const float*)d_in[0];
    const int*   labels = (const int*)d_in[1];
    float*       out    = (float*)d_out;
    float*       en     = (float*)d_ws;            // B*256 floats (8 MB)

    const int B = in_sizes[0] / DIMK;              // 8192

    // Kernel 1: normalize (one wave per row, 8 rows/block)
    cl_normalize<<<(B + WAVES - 1) / WAVES, 32 * WAVES, 0, stream>>>(emb, en, out, B);

    // Kernel 2: fused loss (one wave per 16-row tile, 8 waves/block)
    const int tiles  = B / 16;                     // 512
    const int blocks = tiles / WAVES;              // 64
    cl_loss<<<blocks, 32 * WAVES, 0, stream>>>(en, labels, out, B);
}